// FastLinearAttention_62749472194585
// MI455X (gfx1250) — compile-verified
//
#include <hip/hip_runtime.h>

typedef __attribute__((ext_vector_type(2))) float v2f;
typedef __attribute__((ext_vector_type(8))) float v8f;

#define NSEQ  8192
#define DDIM  64
#define FDIM  128
#define NCOL  80              // 64 V cols + 1 ones col (K_sum) + 15 pad
#define KVSZ  (FDIM * NCOL)   // 10240 floats per head
#define CHUNK 512
#define BLK   64
#define NBLK  (CHUNK / BLK)   // 8

// ---- CDNA5 async global->LDS path: inline asm (builtin pointer ASes are not
// spellable from HIP); s_wait_asynccnt via builtin. Falls back to sync copy. ----
#if defined(__has_builtin)
#if __has_builtin(__builtin_amdgcn_s_wait_asynccnt)
#define USE_ASYNC_LDS 1
#endif
#endif
#ifndef USE_ASYNC_LDS
#define USE_ASYNC_LDS 0
#endif

__device__ __forceinline__ void copy_f4(const float4* g, float4* l) {
#if USE_ASYNC_LDS
    // LDS aperture flat address: low 32 bits == wave-relative LDS byte offset
    const unsigned loff = (unsigned)(size_t)l;
    asm volatile("global_load_async_to_lds_b128 %0, %1, off"
                 :: "v"(loff), "v"(g)
                 : "memory");
#else
    *l = *g;
#endif
}

#if USE_ASYNC_LDS
#define WAIT_ASYNC(n) __builtin_amdgcn_s_wait_asynccnt(n)
#else
#define WAIT_ASYNC(n)
#endif

// ---------------- kernel 0: zero the KV workspace ----------------
__global__ void FLA_zero_ws(float4* __restrict__ ws) {
    ws[blockIdx.x * 256 + threadIdx.x] = make_float4(0.f, 0.f, 0.f, 0.f);
}

// ---------------- kernel 1: KVext = phi(K)^T @ [V | 1 | 0pad] ----------------
__global__ void __launch_bounds__(256)
FLA_pass1_kv(const float* __restrict__ K, const float* __restrict__ V,
             float* __restrict__ ws) {
    const int bh    = blockIdx.x >> 4;   // 64 heads
    const int chunk = blockIdx.x & 15;   // 16 sequence chunks of 512 rows
    const int tid   = threadIdx.x;
    const int w     = tid >> 5;          // wave id 0..7
    const int lane  = tid & 31;
    const int li    = lane & 15;
    const int half  = lane >> 4;

    // double-buffered stage: 2 x (16KB K + 16KB V) = 64KB LDS
    __shared__ __align__(16) float ldsK[2][BLK * DDIM];
    __shared__ __align__(16) float ldsV[2][BLK * DDIM];

    const int  m0  = w << 4;                 // feature tile base (0..112)
    const int  d   = (m0 + li) & 63;         // source K column
    const bool neg = (m0 + li) >= 64;        // second half of phi

    v8f acc[5] = {};                         // 16x16 f32 tiles, 5 N-tiles (80 cols)

    // constant B tile for cols 64..79: col 64 == ones (produces K_sum)
    v2f b4; b4.x = (li == 0) ? 1.0f : 0.0f; b4.y = b4.x;

    const long   hb = (long)bh * NSEQ * DDIM;
    const float* Kc = K + hb + (long)chunk * CHUNK * DDIM;
    const float* Vc = V + hb + (long)chunk * CHUNK * DDIM;

    // ---- issue stage 0 (8 async b128 per thread: 4 K + 4 V) ----
    {
        const float4* Kg = (const float4*)Kc;
        const float4* Vg = (const float4*)Vc;
        #pragma unroll
        for (int i = 0; i < 4; ++i) {
            copy_f4(Kg + tid + i * 256, (float4*)ldsK[0] + tid + i * 256);
            copy_f4(Vg + tid + i * 256, (float4*)ldsV[0] + tid + i * 256);
        }
    }

    for (int nb = 0; nb < NBLK; ++nb) {
        const int cur = nb & 1;

        if (nb + 1 < NBLK) {
            // issue next stage into the other buffer, then wait for current
            const float4* Kg = (const float4*)(Kc + (nb + 1) * BLK * DDIM);
            const float4* Vg = (const float4*)(Vc + (nb + 1) * BLK * DDIM);
            #pragma unroll
            for (int i = 0; i < 4; ++i) {
                copy_f4(Kg + tid + i * 256, (float4*)ldsK[cur ^ 1] + tid + i * 256);
                copy_f4(Vg + tid + i * 256, (float4*)ldsV[cur ^ 1] + tid + i * 256);
            }
#if !USE_ASYNC_LDS
            __builtin_prefetch(Kc + (nb + 1) * BLK * DDIM + tid * 16, 0, 0);
#endif
            WAIT_ASYNC(8);      // 8 newer loads may stay in flight; current stage done
        } else {
            WAIT_ASYNC(0);
        }
        __syncthreads();

        const float* lK = ldsK[cur];
        const float* lV = ldsV[cur];

        // ---- 16 WMMA K-steps of 4 sequence rows each ----
        #pragma unroll 4
        for (int k4 = 0; k4 < BLK / 4; ++k4) {
            const int r0 = (k4 << 2) + (half << 1);   // K rows {r0, r0+1}
            float a0 = lK[r0 * DDIM + d];
            float a1 = lK[(r0 + 1) * DDIM + d];
            if (neg) { a0 = -a0; a1 = -a1; }
            v2f a; a.x = fmaxf(a0, 0.0f); a.y = fmaxf(a1, 0.0f);   // phi(K)

            #pragma unroll
            for (int t = 0; t < 4; ++t) {
                v2f b;
                b.x = lV[r0 * DDIM + (t << 4) + li];
                b.y = lV[(r0 + 1) * DDIM + (t << 4) + li];
                acc[t] = __builtin_amdgcn_wmma_f32_16x16x4_f32(
                    false, a, false, b, (short)0, acc[t], false, false);
            }
            acc[4] = __builtin_amdgcn_wmma_f32_16x16x4_f32(
                false, a, false, b4, (short)0, acc[4], false, false);
        }
        __syncthreads();   // protect 'cur' buffer before it is re-staged
    }

    // ---- merge chunk partials: ws[bh][f][c] += acc ----
    float* wsb = ws + (long)bh * KVSZ;
    #pragma unroll
    for (int t = 0; t < 5; ++t) {
        #pragma unroll
        for (int j = 0; j < 8; ++j) {
            const int row = m0 + j + (half << 3);     // C/D layout: VGPR j -> M=j / M=8+j
            const int col = (t << 4) + li;
            atomicAdd(&wsb[row * NCOL + col], acc[t][j]);
        }
    }
}

// ---------------- kernel 2: out = (phi(Q) @ KVext) with fused normalizer ----------------
__global__ void __launch_bounds__(256)
FLA_pass2_out(const float* __restrict__ Q, const float* __restrict__ ws,
              float* __restrict__ out) {
    const int bh   = blockIdx.x >> 6;    // 64 heads
    const int rg   = blockIdx.x & 63;    // 64 row-groups of 128 rows
    const int tid  = threadIdx.x;
    const int w    = tid >> 5;
    const int lane = tid & 31;
    const int li   = lane & 15;
    const int half = lane >> 4;

    __shared__ __align__(16) float ldsKV[KVSZ];  // 40 KB: whole KVext of this head

    const float4* wsg = (const float4*)(ws + (long)bh * KVSZ);
    float4* kl = (float4*)ldsKV;
    #pragma unroll
    for (int i = 0; i < KVSZ / 4 / 256; ++i)     // 10 b128 per thread
        copy_f4(wsg + tid + i * 256, kl + tid + i * 256);
    WAIT_ASYNC(0);
    __syncthreads();

    const int    n0   = rg * 128 + (w << 4);     // this wave's 16 output rows
    const float* Qrow = Q + ((long)bh * NSEQ + n0 + li) * DDIM;
    __builtin_prefetch(Qrow, 0, 0);

    v8f acc[5] = {};

    #pragma unroll 4
    for (int k0 = 0; k0 < FDIM; k0 += 4) {
        const int kk = k0 + (half << 1);         // this half's K indices {kk, kk+1}
        const int dq = kk & 63;
        float a0 = Qrow[dq];
        float a1 = Qrow[dq + 1];
        if (kk >= 64) { a0 = -a0; a1 = -a1; }
        v2f a; a.x = fmaxf(a0, 0.0f); a.y = fmaxf(a1, 0.0f);   // phi(Q)

        #pragma unroll
        for (int t = 0; t < 5; ++t) {
            v2f b;
            b.x = ldsKV[kk * NCOL + (t << 4) + li];
            b.y = ldsKV[(kk + 1) * NCOL + (t << 4) + li];
            acc[t] = __builtin_amdgcn_wmma_f32_16x16x4_f32(
                false, a, false, b, (short)0, acc[t], false, false);
        }
    }

    // normalizer lives in tile 4, column 64 (lane 0 / lane 16 of each VGPR)
    float* ob = out + ((long)bh * NSEQ + n0) * DDIM;
    #pragma unroll
    for (int j = 0; j < 8; ++j) {
        const float nrm = __shfl(acc[4][j], lane & 16, 32) + 1e-6f;
        const float inv = 1.0f / nrm;
        const int   row = j + (half << 3);
        #pragma unroll
        for (int t = 0; t < 4; ++t)
            ob[row * DDIM + (t << 4) + li] = acc[t][j] * inv;
    }
}

extern "C" void kernel_launch(void* const* d_in, const int* in_sizes, int n_in,
                              void* d_out, int out_size, void* d_ws, size_t ws_size,
                              hipStream_t stream) {
    const float* Q = (const float*)d_in[0];
    const float* K = (const float*)d_in[1];
    const float* V = (const float*)d_in[2];
    float* out = (float*)d_out;
    float* ws  = (float*)d_ws;   // needs 64*128*80*4 = 2.62 MB

    // 64*KVSZ floats = 163840 float4 = 640 blocks * 256 threads
    FLA_zero_ws<<<640, 256, 0, stream>>>((float4*)ws);
    FLA_pass1_kv<<<64 * 16, 256, 0, stream>>>(K, V, ws);
    FLA_pass2_out<<<64 * 64, 256, 0, stream>>>(Q, ws, out);
}